// LocalAttention_35064113005055
// MI455X (gfx1250) — compile-verified
//
#include <hip/hip_runtime.h>
#include <math.h>

typedef __attribute__((ext_vector_type(2))) float v2f;
typedef __attribute__((ext_vector_type(8))) float v8f;
typedef int v4i __attribute__((vector_size(16)));

#define HID   1024
#define WIN   129
#define DHALF 64
#define BATCH 128
#define SLEN  2048
#define ROWS  (BATCH * WIN)   // 16512 = 64 * 258

// GEMM tiling
#define KC    64              // K chunk staged in LDS
#define APAD  68              // LDS row stride (floats) for A: 68 % 64 == 4 -> conflict-free
#define BPAD  264             // LDS row stride (floats) for B: half-wave bank sets disjoint

#if __has_builtin(__builtin_amdgcn_global_load_async_to_lds_b128) && \
    __has_builtin(__builtin_amdgcn_s_wait_asynccnt)
#define USE_ASYNC_LDS 1
#else
#define USE_ASYNC_LDS 0
#endif

#if USE_ASYNC_LDS
__device__ __forceinline__ void cp16_async(float* ldst, const float* gsrc) {
    __builtin_amdgcn_global_load_async_to_lds_b128(
        (__attribute__((address_space(1))) v4i*)(unsigned long long)(size_t)gsrc,
        (__attribute__((address_space(3))) v4i*)(unsigned)(size_t)ldst,
        0, 0);
}
#endif

// ---------------------------------------------------------------------------
// Kernel 1: predictive position  pt[b], base[b]
//   sigmoid_inp = tanh(s @ W_omega_w + b_omega_w) @ W_omega_v + b_omega_v
//   pt = (SLEN - 2D - 1) * sigmoid(.) + D ; base = floor(pt)
// ---------------------------------------------------------------------------
__global__ __launch_bounds__(256) void pt_kernel(
    const float* __restrict__ s,
    const float* __restrict__ Wow, const float* __restrict__ bow,
    const float* __restrict__ Wov, const float* __restrict__ bov,
    float* __restrict__ pt_out, int* __restrict__ base_out)
{
    __shared__ float sl[HID];
    __shared__ float red[256];
    const int b = blockIdx.x;
    const int tid = threadIdx.x;

    *(float4*)&sl[tid * 4] = *(const float4*)(s + (size_t)b * HID + tid * 4);
    __syncthreads();

    float part = 0.f;
    for (int j = 0; j < 4; ++j) {
        const int h2 = j * 256 + tid;
        float acc = bow[h2];
        const float* wcol = Wow + h2;
#pragma unroll 4
        for (int h1 = 0; h1 < HID; ++h1)
            acc = fmaf(sl[h1], wcol[(size_t)h1 * HID], acc);
        part += tanhf(acc) * Wov[h2];
    }
    red[tid] = part;
    __syncthreads();
    for (int st = 128; st > 0; st >>= 1) {
        if (tid < st) red[tid] += red[tid + st];
        __syncthreads();
    }
    if (tid == 0) {
        const float v  = red[0] + bov[0];
        const float sg = 1.f / (1.f + expf(-v));
        const float pt = (float)(SLEN - 2 * DHALF - 1) * sg + (float)DHALF;
        pt_out[b]   = pt;
        base_out[b] = (int)floorf(pt);
    }
}

// ---------------------------------------------------------------------------
// Kernel 2: enc_w_t[r, :] = enc_output[b, base[b]+w-D, :] @ W_attn + b_attn
// Block = 256 thr (8 wave32) computing a 64x256 tile; waves in 2(M) x 4(N).
// Each wave: 32 rows x 64 cols = 8 f32 16x16 accum tiles.
// K staged in LDS chunks of 64 via GLOBAL_LOAD_ASYNC_TO_LDS_B128 (ASYNCcnt).
// ---------------------------------------------------------------------------
__global__ __launch_bounds__(256) void attn_gemm_kernel(
    const float* __restrict__ enc,
    const float* __restrict__ Wattn,
    const float* __restrict__ battn,
    const int*   __restrict__ basep,
    float* __restrict__ out_encwt)
{
    __shared__ float lA[64 * APAD];   // 17408 B
    __shared__ float lB[KC * BPAD];   // 67584 B
    const int tid  = threadIdx.x;
    const int rt   = blockIdx.x;             // 0..257 row tiles (64 rows each)
    const int nblk = blockIdx.y * 256;       // 0..3  -> col offset

    // ---- per-thread staging assignments ----
    const int rloc0 = tid >> 4;              // A: row 0..15 (+16 per pass)
    const int aseg  = (tid & 15) * 4;        // A: float4 segment within row
    const int bkr0  = tid >> 6;              // B: k-row 0..3 (+4 per pass)
    const int bseg  = (tid & 63) * 4;        // B: float4 segment within row

    // precompute gathered A row pointers (window gather fused here)
    const float* gA[4];
#pragma unroll
    for (int p = 0; p < 4; ++p) {
        const int row  = p * 16 + rloc0;     // 0..63
        const int r    = rt * 64 + row;
        const int b    = r / WIN;
        const int w    = r - b * WIN;
        const int srow = basep[b] + w - DHALF;      // in-range by construction
        gA[p] = enc + ((size_t)b * SLEN + srow) * HID + aseg;
    }

    const int wave = tid >> 5;
    const int lane = tid & 31;
    const int half = lane >> 4;              // 0: K=k,k+1   1: K=k+2,k+3
    const int l16  = lane & 15;
    const int mw   = wave >> 2;              // 0..1 -> row offset 0/32
    const int nw   = wave & 3;               // 0..3 -> col offset 0/64/128/192
    const int mrow = mw * 32;
    const int ncol = nw * 64 + l16;          // LDS-local column

    v8f c0[4] = {{}, {}, {}, {}};            // rows mrow..mrow+15
    v8f c1[4] = {{}, {}, {}, {}};            // rows mrow+16..mrow+31

    for (int k0 = 0; k0 < HID; k0 += KC) {
        __syncthreads();                      // previous chunk consumed
#if USE_ASYNC_LDS
#pragma unroll
        for (int p = 0; p < 4; ++p)
            cp16_async(&lA[(p * 16 + rloc0) * APAD + aseg], gA[p] + k0);
#pragma unroll
        for (int p = 0; p < 16; ++p) {
            const int kr = p * 4 + bkr0;
            cp16_async(&lB[kr * BPAD + bseg],
                       Wattn + (size_t)(k0 + kr) * HID + nblk + bseg);
        }
        __builtin_amdgcn_s_wait_asynccnt(0);
#else
#pragma unroll
        for (int p = 0; p < 4; ++p) {
            float4 v = *(const float4*)(gA[p] + k0);
            *(float4*)&lA[(p * 16 + rloc0) * APAD + aseg] = v;
        }
#pragma unroll
        for (int p = 0; p < 16; ++p) {
            const int kr = p * 4 + bkr0;
            float4 v = *(const float4*)(Wattn + (size_t)(k0 + kr) * HID + nblk + bseg);
            *(float4*)&lB[kr * BPAD + bseg] = v;
        }
#endif
        __syncthreads();

#pragma unroll 8
        for (int kk = 0; kk < KC; kk += 4) {
            const int ka = kk + 2 * half;
            // A fragments: lanes0-15 -> (row, K=k..k+1); lanes16-31 -> K=k+2..k+3
            v2f a0 = *(const v2f*)&lA[(mrow + l16) * APAD + ka];
            v2f a1 = *(const v2f*)&lA[(mrow + 16 + l16) * APAD + ka];
#pragma unroll
            for (int t = 0; t < 4; ++t) {
                v2f bf;
                bf.x = lB[ka * BPAD + ncol + t * 16];
                bf.y = lB[(ka + 1) * BPAD + ncol + t * 16];
                c0[t] = __builtin_amdgcn_wmma_f32_16x16x4_f32(false, a0, false, bf, (short)0, c0[t], false, false);
                c1[t] = __builtin_amdgcn_wmma_f32_16x16x4_f32(false, a1, false, bf, (short)0, c1[t], false, false);
            }
        }
    }

    // ---- epilogue: C/D layout -> lanes0-15 M=i, lanes16-31 M=i+8, N=lane ----
    const int rb0 = rt * 64 + mrow + half * 8;
#pragma unroll
    for (int t = 0; t < 4; ++t) {
        const int col = nblk + nw * 64 + t * 16 + l16;
        const float bb = battn[col];
#pragma unroll
        for (int i = 0; i < 8; ++i) {
            out_encwt[(size_t)(rb0 + i) * HID + col]      = c0[t][i] + bb;
            out_encwt[(size_t)(rb0 + 16 + i) * HID + col] = c1[t][i] + bb;
        }
    }
}

// ---------------------------------------------------------------------------
// Kernel 3: alpha[r] = dot(enc_w_t[r,:], s[b,:])  -- one wave per row
// ---------------------------------------------------------------------------
__global__ __launch_bounds__(256) void alpha_kernel(
    const float* __restrict__ encwt,
    const float* __restrict__ s,
    float* __restrict__ alpha)
{
    const int wave = threadIdx.x >> 5;
    const int lane = threadIdx.x & 31;
    const int r = blockIdx.x * 8 + wave;      // < 16512
    const int b = r / WIN;
    const float* e  = encwt + (size_t)r * HID;
    const float* sp = s + (size_t)b * HID;
    float acc = 0.f;
#pragma unroll
    for (int j = 0; j < 8; ++j) {
        const int off = (j * 32 + lane) * 4;
        float4 ev = *(const float4*)(e + off);
        float4 sv = *(const float4*)(sp + off);
        acc = fmaf(ev.x, sv.x, acc);
        acc = fmaf(ev.y, sv.y, acc);
        acc = fmaf(ev.z, sv.z, acc);
        acc = fmaf(ev.w, sv.w, acc);
    }
    for (int m = 16; m > 0; m >>= 1) acc += __shfl_xor(acc, m, 32);
    if (lane == 0) alpha[r] = acc;
}

// ---------------------------------------------------------------------------
// Kernel 4: softmax over w of alpha[b,w] * exp(-((base+w-D)-pt)^2 / 2048)
// One wave per batch sample.
// ---------------------------------------------------------------------------
__global__ __launch_bounds__(32) void softmax_kernel(
    const float* __restrict__ alpha,
    const float* __restrict__ ptv,
    const int*   __restrict__ basev,
    float* __restrict__ out)
{
    const int b = blockIdx.x;
    const int lane = threadIdx.x;
    const float pt = ptv[b];
    const int base = basev[b];

    float e[5];
    float mx = -1e30f;
#pragma unroll
    for (int j = 0; j < 5; ++j) {
        const int w = j * 32 + lane;
        if (w < WIN) {
            const float d = (float)(base + w - DHALF) - pt;
            const float g = expf(-(d * d) * (1.0f / 2048.0f));
            e[j] = alpha[b * WIN + w] * g;
            mx = fmaxf(mx, e[j]);
        } else {
            e[j] = -1e30f;
        }
    }
    for (int m = 16; m > 0; m >>= 1) mx = fmaxf(mx, __shfl_xor(mx, m, 32));
    float sum = 0.f;
#pragma unroll
    for (int j = 0; j < 5; ++j) {
        e[j] = (e[j] > -1e29f) ? expf(e[j] - mx) : 0.f;
        sum += e[j];
    }
    for (int m = 16; m > 0; m >>= 1) sum += __shfl_xor(sum, m, 32);
    const float inv = 1.f / sum;
#pragma unroll
    for (int j = 0; j < 5; ++j) {
        const int w = j * 32 + lane;
        if (w < WIN) out[b * WIN + w] = e[j] * inv;
    }
}

// ---------------------------------------------------------------------------
extern "C" void kernel_launch(void* const* d_in, const int* in_sizes, int n_in,
                              void* d_out, int out_size, void* d_ws, size_t ws_size,
                              hipStream_t stream)
{
    const float* enc   = (const float*)d_in[0];   // [B, SLEN, H]
    const float* s     = (const float*)d_in[1];   // [B, H]
    const float* Wow   = (const float*)d_in[2];   // [H, H]
    const float* bow   = (const float*)d_in[3];   // [H]
    const float* Wov   = (const float*)d_in[4];   // [H, 1]
    const float* bov   = (const float*)d_in[5];   // [1]
    const float* Wattn = (const float*)d_in[6];   // [H, H]
    const float* battn = (const float*)d_in[7];   // [H]

    float* out       = (float*)d_out;
    float* out_sm    = out;                        // [B*WIN]
    float* out_encwt = out + (size_t)BATCH * WIN;  // [B*WIN*H]

    char*  ws    = (char*)d_ws;
    float* pt    = (float*)ws;                     // B floats
    int*   base  = (int*)(ws + 512);               // B ints
    float* alpha = (float*)(ws + 1024);            // B*WIN floats

    pt_kernel<<<BATCH, 256, 0, stream>>>(s, Wow, bow, Wov, bov, pt, base);

    dim3 gg(ROWS / 64, HID / 256);                 // 258 x 4
    attn_gemm_kernel<<<gg, 256, 0, stream>>>(enc, Wattn, battn, base, out_encwt);

    alpha_kernel<<<ROWS / 8, 256, 0, stream>>>(out_encwt, s, alpha);

    softmax_kernel<<<BATCH, 32, 0, stream>>>(alpha, pt, base, out_sm);
}